// DiseaseHead_59674275610776
// MI455X (gfx1250) — compile-verified
//
#include <hip/hip_runtime.h>
#include <math.h>

typedef __attribute__((ext_vector_type(2))) float v2f;
typedef __attribute__((ext_vector_type(4))) float v4f;
typedef __attribute__((ext_vector_type(8))) float v8f;

#define B_  8192
#define P_  25
#define C_  256
#define D_  5
#define R_  (B_ * P_)            // 204800 rows
#define ROWS_PER_WAVE 16
#define WAVES_PER_BLOCK 2
#define HSTRIDE 260              // 260 % 64 == 4 -> conflict-free b64 A/B loads
#define KSTEPS (C_ / 4)          // 64 chained wmma f32 16x16x4 steps
#define ZOFF (D_ * HSTRIDE)      // zero slot for padded B columns

// ---------------------------------------------------------------------------
// Kernel 1: w_rowsum flat row r = (p*256+c): sum of 256 contiguous floats.
// One wave per row, 6400 waves over 800 blocks -> saturates HBM for 6.5 MB.
// ---------------------------------------------------------------------------
__global__ __launch_bounds__(256) void rowsum_kernel(const float* __restrict__ w,
                                                     float* __restrict__ wrow) {
  const int row = blockIdx.x * 8 + (threadIdx.x >> 5);   // 0..6399
  const int ln  = threadIdx.x & 31;
  const float* rp = w + (size_t)row * C_;
  float s = 0.f;
#pragma unroll
  for (int j = 0; j < 8; ++j) s += rp[ln + 32 * j];
#pragma unroll
  for (int off = 16; off > 0; off >>= 1) s += __shfl_xor(s, off, 32);
  if (ln == 0) wrow[row] = s;
}

// ---------------------------------------------------------------------------
// Kernel 2: fused pointwise + LayerNorm + residual/ReLU + WMMA GEMV.
// Wave w owns p = w%25 and 16 batch rows b0..b0+15 (b0 = (w/25)*16), so the
// per-point tensors (w_rowsum, bias) and gamma/beta are loaded ONCE per wave.
// ---------------------------------------------------------------------------
__global__ __launch_bounds__(64) void fused_kernel(
    const float* __restrict__ feat, const float* __restrict__ wrow,
    const float* __restrict__ bias, const float* __restrict__ gamma,
    const float* __restrict__ beta, const float* __restrict__ linw,
    const float* __restrict__ linb, float* __restrict__ out) {
  __shared__ __attribute__((aligned(16))) float hbuf[WAVES_PER_BLOCK * ROWS_PER_WAVE * HSTRIDE];
  __shared__ __attribute__((aligned(16))) float lw[ZOFF + 4];  // padded lin_w + zero slot

  const int tid = threadIdx.x;

  // Stage lin_w [5][256] into LDS with stride 260 (bank-conflict-free reads),
  // plus a 16-byte zero slot that padded B columns (5..15) broadcast-read.
  for (int e = tid; e < D_ * C_; e += 64) {
    lw[(e >> 8) * HSTRIDE + (e & 255)] = linw[e];
  }
  if (tid < 4) lw[ZOFF + tid] = 0.f;
  __syncthreads();

  const int wv   = tid >> 5;
  const int ln   = tid & 31;
  const int wave = blockIdx.x * WAVES_PER_BLOCK + wv;    // 0..12799
  const int p    = wave % P_;
  const int b0   = (wave / P_) * ROWS_PER_WAVE;
  float* hw = hbuf + wv * (ROWS_PER_WAVE * HSTRIDE);

  // Lane owns columns {4*ln..4*ln+3} and {4*ln+128..4*ln+131}.
  const int c0 = ln * 4, c1 = ln * 4 + 128;

  // Hoisted per-wave loads (p fixed, reused for all 16 rows).
  const v4f wr0 = *(const v4f*)(wrow + p * C_ + c0);
  const v4f wr1 = *(const v4f*)(wrow + p * C_ + c1);
  const v4f bi0 = *(const v4f*)(bias + p * C_ + c0);
  const v4f bi1 = *(const v4f*)(bias + p * C_ + c1);
  const v4f ga0 = *(const v4f*)(gamma + c0);
  const v4f ga1 = *(const v4f*)(gamma + c1);
  const v4f be0 = *(const v4f*)(beta + c0);
  const v4f be1 = *(const v4f*)(beta + c1);

  // ---- Phase A: per-row pointwise + LayerNorm + residual ReLU -> LDS ----
#pragma unroll 2
  for (int r = 0; r < ROWS_PER_WAVE; ++r) {
    const int g = (b0 + r) * P_ + p;
    const float* fp = feat + (size_t)g * C_;
    const v4f f0 = __builtin_nontemporal_load((const v4f*)(fp + c0));
    const v4f f1 = __builtin_nontemporal_load((const v4f*)(fp + c1));
    v4f pw0, pw1;
    float s1 = 0.f, s2 = 0.f;
#pragma unroll
    for (int q = 0; q < 4; ++q) {
      pw0[q] = fmaf(f0[q], wr0[q], bi0[q]);
      pw1[q] = fmaf(f1[q], wr1[q], bi1[q]);
      s1 += pw0[q] + pw1[q];
      s2  = fmaf(pw0[q], pw0[q], s2);
      s2  = fmaf(pw1[q], pw1[q], s2);
    }
#pragma unroll
    for (int off = 16; off > 0; off >>= 1) {
      s1 += __shfl_xor(s1, off, 32);
      s2 += __shfl_xor(s2, off, 32);
    }
    const float inv  = 1.0f / (float)C_;
    const float mean = s1 * inv;
    const float var  = fmaf(s2, inv, -mean * mean);
    const float rstd = rsqrtf(var + 1e-5f);
    v4f h0, h1;
#pragma unroll
    for (int q = 0; q < 4; ++q) {
      const float n0 = fmaf((pw0[q] - mean) * rstd, ga0[q], be0[q]);
      const float n1 = fmaf((pw1[q] - mean) * rstd, ga1[q], be1[q]);
      h0[q] = fmaxf(f0[q] + n0, 0.f);
      h1[q] = fmaxf(f1[q] + n1, 0.f);
    }
    *(v4f*)&hw[r * HSTRIDE + c0] = h0;
    *(v4f*)&hw[r * HSTRIDE + c1] = h1;
  }

  // ---- Phase B: out[16 rows][5] = h[16][256] x lin_w^T via chained WMMA ----
  // A fragment (16x4 f32): lane<16 -> row=lane, K={4k,4k+1};
  //                        lane>=16 -> row=lane-16, K={4k+2,4k+3}.
  // B fragment: lane reads lin_w[col][4k+koff .. +1]; padded cols read zeros.
  const int rowA = ln & 15;
  const int koff = (ln >> 4) << 1;
  const int dcol = ln & 15;
  const float* bbase = (dcol < D_) ? &lw[dcol * HSTRIDE + koff] : &lw[ZOFF];
  const int bstep = (dcol < D_) ? 4 : 0;

  v8f acc = {0.f, 0.f, 0.f, 0.f, 0.f, 0.f, 0.f, 0.f};
#pragma unroll 4
  for (int k = 0; k < KSTEPS; ++k) {
    v2f a = *(const v2f*)&hw[rowA * HSTRIDE + 4 * k + koff];
    v2f b = *(const v2f*)&bbase[k * bstep];
    acc = __builtin_amdgcn_wmma_f32_16x16x4_f32(
        /*neg_a=*/false, a, /*neg_b=*/false, b,
        /*c_mod=*/(short)0, acc, /*reuse_a=*/false, /*reuse_b=*/false);
  }

  // D layout: VGPR j -> row j (lanes 0-15) / row j+8 (lanes 16-31), N = ln&15.
  const int rAdd = (ln >> 4) << 3;
  if (dcol < D_) {
    const float lb = linb[dcol];
#pragma unroll
    for (int j = 0; j < 8; ++j) {
      const int g = (b0 + j + rAdd) * P_ + p;
      __builtin_nontemporal_store(acc[j] + lb, out + (size_t)g * D_ + dcol);
    }
  }
}

// ---------------------------------------------------------------------------
extern "C" void kernel_launch(void* const* d_in, const int* in_sizes, int n_in,
                              void* d_out, int out_size, void* d_ws, size_t ws_size,
                              hipStream_t stream) {
  const float* feat    = (const float*)d_in[0];
  const float* weights = (const float*)d_in[1];
  const float* bias    = (const float*)d_in[2];
  const float* gamma   = (const float*)d_in[3];
  const float* beta    = (const float*)d_in[4];
  const float* linw    = (const float*)d_in[5];
  const float* linb    = (const float*)d_in[6];
  float* out  = (float*)d_out;
  float* wrow = (float*)d_ws;   // 25*256 floats = 25,600 B scratch

  rowsum_kernel<<<(P_ * C_) / 8, 256, 0, stream>>>(weights, wrow);

  const int nblocks = R_ / (ROWS_PER_WAVE * WAVES_PER_BLOCK);  // 6400, exact
  fused_kernel<<<nblocks, 64, 0, stream>>>(feat, wrow, bias, gamma, beta,
                                           linw, linb, out);
}